// ChamferLoss_51110110823173
// MI455X (gfx1250) — compile-verified
//
#include <hip/hip_runtime.h>
#include <hip/hip_bf16.h>

typedef __attribute__((ext_vector_type(2))) float v2f;
typedef __attribute__((ext_vector_type(8))) float v8f;

#define NPTS   8192
#define CHUNK  256          // rows staged in LDS per stage (4 KB as float4)
#define TILES_PER_BATCH (NPTS / 16)   // 512 column tiles of 16

// For each "column" point y_m (kept dimension), compute min over all "row"
// points x_n of ||x_n - y_m||^2, then per-wave sum of its 16 column-mins.
// partial[b*512 + tile] = sum of 16 column mins.
__global__ __launch_bounds__(256) void chamfer_min_wmma(
    const float* __restrict__ x,   // [B, NPTS, 3]  reduced-over set (A rows)
    const float* __restrict__ y,   // [B, NPTS, 3]  kept set (B cols)
    float* __restrict__ partial)   // [B, 512]
{
    const int b     = blockIdx.x;
    const int lane  = threadIdx.x & 31;
    const int wave  = threadIdx.x >> 5;
    const int col16 = lane & 15;
    const bool hi   = lane >= 16;

    const int tile = blockIdx.y * 8 + wave;     // 0..511
    const int col  = tile * 16 + col16;

    const float* xb = x + (size_t)b * NPTS * 3;
    const float* yb = y + (size_t)b * NPTS * 3;

    // ---- loop-invariant B matrix (4x16) and C accumulator (||y||^2 per col)
    const float y0 = yb[col * 3 + 0];
    const float y1 = yb[col * 3 + 1];
    const float y2 = yb[col * 3 + 2];
    const float ynrm = y0 * y0 + y1 * y1 + y2 * y2;

    v2f bmat;
    bmat.x = hi ? (-2.0f * y2) : (-2.0f * y0);   // K=2 | K=0
    bmat.y = hi ? 1.0f         : (-2.0f * y1);   // K=3 | K=1

    v8f cvec;
#pragma unroll
    for (int i = 0; i < 8; ++i) cvec[i] = ynrm;  // C[n,m] = ||y_m||^2

    // LDS tile: float4 {x0, x1, x2, ||x||^2} per row.
    // Each lane only ever reads the 8-byte half it feeds into A:
    //   lanes 0-15  -> {x0, x1}      (K=0,1)
    //   lanes 16-31 -> {x2, ||x||^2} (K=2,3)
    __shared__ float4 lds[CHUNK];
    const float2* lds2 = (const float2*)lds;
    const int lbase = col16 * 2 + (hi ? 1 : 0);  // float2 index within a 16-row tile

    // 4 independent min accumulators -> dependency chain depth / 4
    float m0 = 3.402823466e+38f;
    float m1 = 3.402823466e+38f;
    float m2 = 3.402823466e+38f;
    float m3 = 3.402823466e+38f;

    for (int base = 0; base < NPTS; base += CHUNK) {
        __syncthreads();
        {   // 256 threads cooperatively stage 256 rows, precomputing norms
            const int r = base + threadIdx.x;
            const float a0 = xb[r * 3 + 0];
            const float a1 = xb[r * 3 + 1];
            const float a2 = xb[r * 3 + 2];
            lds[threadIdx.x] = make_float4(a0, a1, a2, a0*a0 + a1*a1 + a2*a2);
        }
        __syncthreads();

#pragma unroll
        for (int t = 0; t < CHUNK / 16; ++t) {
            const float2 a2 = lds2[t * 32 + lbase];  // ds_load_b64 (pairs -> 2addr)

            v2f amat;
            amat.x = a2.x;
            amat.y = a2.y;

            // D = A*B + C : full 16x16 squared-distance tile in one WMMA
            v8f d = __builtin_amdgcn_wmma_f32_16x16x4_f32(
                false, amat, false, bmat, (short)0, cvec, false, false);

            // 4 independent v_min3_num_f32 per WMMA (no serial chain)
            m0 = fminf(m0, fminf(d[0], d[1]));
            m1 = fminf(m1, fminf(d[2], d[3]));
            m2 = fminf(m2, fminf(d[4], d[5]));
            m3 = fminf(m3, fminf(d[6], d[7]));
        }
    }

    float colmin = fminf(fminf(m0, m1), fminf(m2, m3));

    // merge the two half-lane row groups (same column N = lane%16)
    colmin = fminf(colmin, __shfl_xor(colmin, 16, 32));

    // sum the 16 column mins (within each 16-lane group; symmetric across halves)
    float s = colmin;
#pragma unroll
    for (int o = 1; o < 16; o <<= 1)
        s += __shfl_xor(s, o, 32);

    if (lane == 0)
        partial[b * TILES_PER_BATCH + tile] = s;
}

// Deterministic tree reduction: 16 blocks -> d_out[0..7]=loss_1, [8..15]=loss_2
__global__ __launch_bounds__(256) void chamfer_reduce(
    const float* __restrict__ ws, float* __restrict__ out)
{
    const float* basep = ws + (size_t)blockIdx.x * TILES_PER_BATCH;
    __shared__ float s[256];
    float v = basep[threadIdx.x] + basep[threadIdx.x + 256];
    s[threadIdx.x] = v;
    __syncthreads();
#pragma unroll
    for (int o = 128; o > 0; o >>= 1) {
        if (threadIdx.x < o) s[threadIdx.x] += s[threadIdx.x + o];
        __syncthreads();
    }
    if (threadIdx.x == 0)
        out[blockIdx.x] = s[0] * (1.0f / (float)NPTS);
}

extern "C" void kernel_launch(void* const* d_in, const int* in_sizes, int n_in,
                              void* d_out, int out_size, void* d_ws, size_t ws_size,
                              hipStream_t stream) {
    const float* preds = (const float*)d_in[0];   // [8, 8192, 3]
    const float* gts   = (const float*)d_in[1];   // [8, 8192, 3]
    float* out = (float*)d_out;                   // [16] = loss_1(8) ++ loss_2(8)
    float* ws  = (float*)d_ws;                    // 2 * 8 * 512 floats = 32 KB

    dim3 grid(8, NPTS / (8 * 16));   // 8 batches x 64 column blocks
    dim3 block(256);

    // loss_1: min over gts (rows) for each pred (col)
    chamfer_min_wmma<<<grid, block, 0, stream>>>(gts, preds, ws);
    // loss_2: min over preds (rows) for each gt (col)
    chamfer_min_wmma<<<grid, block, 0, stream>>>(preds, gts, ws + 8 * TILES_PER_BATCH);

    chamfer_reduce<<<16, 256, 0, stream>>>(ws, out);
}